// HierarchicalDynamicFFN_14113262535169
// MI455X (gfx1250) — compile-verified
//
#include <hip/hip_runtime.h>
#include <math.h>

// ---------------- problem constants (from reference) ----------------
constexpr int B      = 8;
constexpr int S      = 2048;
constexpr int D      = 1024;   // d_model
constexpr int NIN    = 4096;   // n_input
constexpr int NPROC  = 2048;   // n_process
constexpr int DR     = 256;    // d_routing
constexpr int H      = 512;    // 2*d_routing
constexpr int KIN    = 2048;   // k_input
constexpr int KPROC  = 1024;   // k_process

typedef __attribute__((ext_vector_type(2))) float v2f;
typedef __attribute__((ext_vector_type(8))) float v8f;
typedef __attribute__((ext_vector_type(4))) int   v4i;

__device__ __forceinline__ float gelu_erf(float x) {
    return 0.5f * x * (1.0f + erff(x * 0.70710678118654752440f));
}

// ---------------- CDNA5 async global->LDS helpers ----------------
#if __has_builtin(__builtin_amdgcn_global_load_async_to_lds_b128) && \
    __has_builtin(__builtin_amdgcn_global_load_async_to_lds_b32)
#define HAS_ASYNC_LDS 1
#else
#define HAS_ASYNC_LDS 0
#endif

__device__ __forceinline__ void async_ld_b128(const float* g, float* l) {
#if HAS_ASYNC_LDS
    __builtin_amdgcn_global_load_async_to_lds_b128(
        (__attribute__((address_space(1))) v4i*)(float*)g,
        (__attribute__((address_space(3))) v4i*)l, 0, 0);
#else
    *reinterpret_cast<float4*>(l) = *reinterpret_cast<const float4*>(g);
#endif
}
__device__ __forceinline__ void async_ld_b32(const float* g, float* l) {
#if HAS_ASYNC_LDS
    __builtin_amdgcn_global_load_async_to_lds_b32(
        (__attribute__((address_space(1))) int*)(float*)g,
        (__attribute__((address_space(3))) int*)l, 0, 0);
#else
    *l = *g;
#endif
}
__device__ __forceinline__ void wait_async_lds() {
#if HAS_ASYNC_LDS
#if __has_builtin(__builtin_amdgcn_s_wait_asynccnt)
    __builtin_amdgcn_s_wait_asynccnt(0);
#else
    asm volatile("s_wait_asynccnt 0" ::: "memory");
#endif
#endif
}

// ============================================================
// Kernel 0: gctx[b][d] = max over s of x[b][s][d]
// ============================================================
__global__ __launch_bounds__(256) void k_gctx(const float* __restrict__ x,
                                              float* __restrict__ gctx) {
    int b = blockIdx.x >> 2;
    int d = ((blockIdx.x & 3) << 8) + threadIdx.x;
    const float* p = x + (size_t)b * S * D + d;
    float m = -INFINITY;
    for (int s = 0; s < S; ++s) m = fmaxf(m, p[(size_t)s * D]);
    gctx[b * D + d] = m;
}

// ============================================================
// Kernel 1: router. One block per batch (1024 threads).
// MLP -> LN -> query -> logits -> softmax -> top-2048 select.
// Writes sel_idx[b][2048] (index order) and sel_w = (1-p)+p.
// ============================================================
__global__ __launch_bounds__(1024) void k_router(
    const float* __restrict__ gctx,
    const float* __restrict__ W1, const float* __restrict__ b1,
    const float* __restrict__ lng, const float* __restrict__ lnb,
    const float* __restrict__ W2, const float* __restrict__ b2,
    const float* __restrict__ nkeys,
    int* __restrict__ sel_idx, float* __restrict__ sel_w) {
    __shared__ float sh_h[H];
    __shared__ float sh_q[DR];
    __shared__ float sh_logit[NIN];
    __shared__ float sh_sort[NIN];
    __shared__ float sh_scan[1024];
    __shared__ float sh_stat[4];

    const int b = blockIdx.x, tid = threadIdx.x;
    const float* g = gctx + b * D;

    // h = GELU(gctx @ W1^T + b1)
    if (tid < H) {
        float s = b1[tid];
        const float* w = W1 + (size_t)tid * D;
        for (int d = 0; d < D; ++d) s += g[d] * w[d];
        sh_h[tid] = gelu_erf(s);
    }
    __syncthreads();
    // LayerNorm stats (two-pass, matches reference)
    if (tid == 0) {
        float su = 0.f;
        for (int i = 0; i < H; ++i) su += sh_h[i];
        float mu = su / (float)H;
        float vq = 0.f;
        for (int i = 0; i < H; ++i) { float d0 = sh_h[i] - mu; vq += d0 * d0; }
        sh_stat[0] = mu;
        sh_stat[1] = rsqrtf(vq / (float)H + 1e-5f);
    }
    __syncthreads();
    if (tid < H)
        sh_h[tid] = (sh_h[tid] - sh_stat[0]) * sh_stat[1] * lng[tid] + lnb[tid];
    __syncthreads();
    // query = h @ W2^T + b2
    if (tid < DR) {
        float s = b2[tid];
        const float* w = W2 + (size_t)tid * H;
        for (int i = 0; i < H; ++i) s += sh_h[i] * w[i];
        sh_q[tid] = s;
    }
    __syncthreads();
    // logits = (query @ nkeys^T) / sqrt(DR)
    for (int n = tid; n < NIN; n += 1024) {
        float s = 0.f;
        const float* kr = nkeys + (size_t)n * DR;
        for (int r = 0; r < DR; ++r) s += sh_q[r] * kr[r];
        s *= 0.0625f;
        sh_logit[n] = s;
        sh_sort[n]  = s;
    }
    __syncthreads();
    // bitonic sort descending on sh_sort (N=4096)
    for (int ksz = 2; ksz <= NIN; ksz <<= 1) {
        for (int j = ksz >> 1; j > 0; j >>= 1) {
            for (int t = tid; t < NIN; t += 1024) {
                int ixj = t ^ j;
                if (ixj > t) {
                    float a = sh_sort[t], c = sh_sort[ixj];
                    bool descBlk = ((t & ksz) == 0);
                    if (descBlk ? (a < c) : (a > c)) { sh_sort[t] = c; sh_sort[ixj] = a; }
                }
            }
            __syncthreads();
        }
    }
    const float thresh = sh_sort[KIN - 1];      // k-th largest
    const float maxl2  = 2.0f * sh_sort[0];     // max of logits/T, T=0.5
    __syncthreads();
    // softmax partition Z over exp(2*l - max)
    float part = 0.f;
    for (int n = tid; n < NIN; n += 1024) part += expf(2.0f * sh_logit[n] - maxl2);
    sh_scan[tid] = part;
    __syncthreads();
    for (int off = 512; off > 0; off >>= 1) {
        if (tid < off) sh_scan[tid] += sh_scan[tid + off];
        __syncthreads();
    }
    const float Zinv = 1.0f / sh_scan[0];
    __syncthreads();

    // ---- top-k selection with exact tie handling (index order) ----
    const int base4 = tid * 4;   // each thread owns 4 contiguous indices
    int my_gt = 0, my_eq = 0;
    for (int u = 0; u < 4; ++u) {
        float l = sh_logit[base4 + u];
        if (l > thresh) my_gt++;
        else if (l == thresh) my_eq++;
    }
    // exclusive scan of eq-counts
    sh_scan[tid] = (float)my_eq; __syncthreads();
    for (int off = 1; off < 1024; off <<= 1) {
        float t = (tid >= off) ? sh_scan[tid - off] : 0.f;
        __syncthreads();
        sh_scan[tid] += t;
        __syncthreads();
    }
    int eq_before = (int)sh_scan[tid] - my_eq;
    __syncthreads();
    // total gt
    sh_scan[tid] = (float)my_gt; __syncthreads();
    for (int off = 512; off > 0; off >>= 1) {
        if (tid < off) sh_scan[tid] += sh_scan[tid + off];
        __syncthreads();
    }
    int total_gt = (int)sh_scan[0];
    __syncthreads();
    int need_eq = KIN - total_gt;
    // flags + selection scan
    int flags[4]; int my_sel = 0; int er = eq_before;
    for (int u = 0; u < 4; ++u) {
        float l = sh_logit[base4 + u];
        int f;
        if (l > thresh) f = 1;
        else if (l == thresh) { f = (er < need_eq) ? 1 : 0; er++; }
        else f = 0;
        flags[u] = f; my_sel += f;
    }
    sh_scan[tid] = (float)my_sel; __syncthreads();
    for (int off = 1; off < 1024; off <<= 1) {
        float t = (tid >= off) ? sh_scan[tid - off] : 0.f;
        __syncthreads();
        sh_scan[tid] += t;
        __syncthreads();
    }
    int pos = (int)sh_scan[tid] - my_sel;
    for (int u = 0; u < 4; ++u) {
        if (flags[u]) {
            int n = base4 + u;
            float p = expf(2.0f * sh_logit[n] - maxl2) * Zinv;
            sel_idx[b * KIN + pos] = n;
            sel_w [b * KIN + pos] = (1.0f - p) + p;   // forward value of STE
            pos++;
        }
    }
}

// ============================================================
// WMMA fragment layouts (V_WMMA_F32_16X16X4_F32):
//   A 16x4: lanes 0-15 -> K={0,1}; lanes 16-31 -> K={2,3}
//   B 4x16: VGPR0 -> K rows {0,2}; VGPR1 -> {1,3} by half-wave
//   C 16x16: VGPR v -> rows v (lanes 0-15) / v+8 (lanes 16-31)
// LDS tile stride 36: multiple of 4 (16B-aligned b128 drops) and
// 36*L mod 64 is distinct for L=0..15 -> conflict-free frag reads.
// ============================================================

// ============================================================
// Kernel 2: stage 1  sel_in[b][s][k'] = GELU(x[b,s,:].P[idx[k']]) * w[k']
// C = A(MxK) * B(NxK)^T, gathered B rows. M=S, N=KIN, Kdim=D.
// Double-buffered LDS, async global->LDS fills overlap WMMA compute.
// ============================================================
__global__ __launch_bounds__(256) void k_stage1(
    const float* __restrict__ x, const float* __restrict__ patt,
    const int* __restrict__ selidx, const float* __restrict__ selw,
    float* __restrict__ sel_in) {
    __shared__ __align__(16) float As[2][128][36];
    __shared__ __align__(16) float Bs[2][128][36];
    const int b  = blockIdx.z;
    const int m0 = blockIdx.x * 128;
    const int n0 = blockIdx.y * 128;
    const int tid = threadIdx.x;
    const int lane = tid & 31, wid = tid >> 5;
    const int wm = wid >> 2, wn = wid & 3;   // 2x4 wave grid
    const int half = lane >> 4, lr = lane & 15;

    const float* gA = x + (size_t)b * S * D;
    const int*   sg = selidx + b * KIN;

    v8f acc[4][2];
    v8f zero = {0.f,0.f,0.f,0.f,0.f,0.f,0.f,0.f};
    for (int i = 0; i < 4; ++i) for (int j = 0; j < 2; ++j) acc[i][j] = zero;

    auto issue = [&](int k0, int buf) {
        #pragma unroll
        for (int i = 0; i < 4; ++i) {
            int idx = tid + 256 * i;
            int row = idx >> 3, cg = idx & 7;
            async_ld_b128(gA + (size_t)(m0 + row) * D + k0 + cg * 4, &As[buf][row][cg * 4]);
            async_ld_b128(patt + (size_t)sg[n0 + row] * D + k0 + cg * 4, &Bs[buf][row][cg * 4]);
        }
    };

    constexpr int NC = D / 32;
    issue(0, 0);
    for (int c = 0; c < NC; ++c) {
        const int cur = c & 1;
        wait_async_lds();
        __syncthreads();
        if (c + 1 < NC) issue((c + 1) * 32, cur ^ 1);
        #pragma unroll
        for (int kk = 0; kk < 32; kk += 4) {
            v2f afr[4], bfr[2];
            #pragma unroll
            for (int mi = 0; mi < 4; ++mi) {
                int m = wm * 64 + mi * 16 + lr;
                afr[mi][0] = As[cur][m][kk + 2*half];
                afr[mi][1] = As[cur][m][kk + 2*half + 1];
            }
            #pragma unroll
            for (int ni = 0; ni < 2; ++ni) {
                int n = wn * 32 + ni * 16 + lr;
                bfr[ni][0] = Bs[cur][n][kk + 2*half];
                bfr[ni][1] = Bs[cur][n][kk + 2*half + 1];
            }
            #pragma unroll
            for (int mi = 0; mi < 4; ++mi)
                #pragma unroll
                for (int ni = 0; ni < 2; ++ni)
                    acc[mi][ni] = __builtin_amdgcn_wmma_f32_16x16x4_f32(
                        false, afr[mi], false, bfr[ni], (short)0, acc[mi][ni], false, false);
        }
    }
    float* gC = sel_in + (size_t)b * S * KIN;
    #pragma unroll
    for (int mi = 0; mi < 4; ++mi)
        #pragma unroll
        for (int ni = 0; ni < 2; ++ni) {
            int ncol = n0 + wn * 32 + ni * 16 + lr;
            float w = selw[b * KIN + ncol];
            #pragma unroll
            for (int v = 0; v < 8; ++v) {
                int mrow = m0 + wm * 64 + mi * 16 + half * 8 + v;
                gC[(size_t)mrow * KIN + ncol] = gelu_erf(acc[mi][ni][v]) * w;
            }
        }
}

// ============================================================
// Kernel 3: stage 2  proc_acts[b][s][p] = GELU(sel_in[b,s,:].Wp_gathered)
// B[p][k'] = pw[p][selidx[k']] (column gather straight into LDS via
// per-lane async b32; pw is 32 MB -> L2 resident). M=S, N=NPROC, Kdim=KIN.
// ============================================================
__global__ __launch_bounds__(256) void k_stage2(
    const float* __restrict__ selin_buf, const float* __restrict__ pw,
    const int* __restrict__ selidx, float* __restrict__ pacts) {
    __shared__ __align__(16) float As[2][128][36];
    __shared__ __align__(16) float Bs[2][128][36];
    const int b  = blockIdx.z;
    const int m0 = blockIdx.x * 128;
    const int n0 = blockIdx.y * 128;
    const int tid = threadIdx.x;
    const int lane = tid & 31, wid = tid >> 5;
    const int wm = wid >> 2, wn = wid & 3;
    const int half = lane >> 4, lr = lane & 15;

    const float* gA = selin_buf + (size_t)b * S * KIN;
    const int*   sg = selidx + b * KIN;

    v8f acc[4][2];
    v8f zero = {0.f,0.f,0.f,0.f,0.f,0.f,0.f,0.f};
    for (int i = 0; i < 4; ++i) for (int j = 0; j < 2; ++j) acc[i][j] = zero;

    auto issue = [&](int k0, int buf) {
        #pragma unroll
        for (int i = 0; i < 4; ++i) {
            int idx = tid + 256 * i;
            int row = idx >> 3, cg = idx & 7;
            async_ld_b128(gA + (size_t)(m0 + row) * KIN + k0 + cg * 4, &As[buf][row][cg * 4]);
        }
        #pragma unroll
        for (int i = 0; i < 16; ++i) {          // gathered columns of pw
            int idx = tid + 256 * i;
            int row = idx >> 5, c = idx & 31;
            async_ld_b32(pw + (size_t)(n0 + row) * NIN + sg[k0 + c], &Bs[buf][row][c]);
        }
    };

    constexpr int NC = KIN / 32;
    issue(0, 0);
    for (int c = 0; c < NC; ++c) {
        const int cur = c & 1;
        wait_async_lds();
        __syncthreads();
        if (c + 1 < NC) issue((c + 1) * 32, cur ^ 1);
        #pragma unroll
        for (int kk = 0; kk < 32; kk += 4) {
            v2f afr[4], bfr[2];
            #pragma unroll
            for (int mi = 0; mi < 4; ++mi) {
                int m = wm * 64 + mi * 16 + lr;
                afr[mi][0] = As[cur][m][kk + 2*half];
                afr[mi][1] = As[cur][m][kk + 2*half + 1];
            }
            #pragma unroll
            for (int ni = 0; ni < 2; ++ni) {
                int n = wn * 32 + ni * 16 + lr;
                bfr[ni][0] = Bs[cur][n][kk + 2*half];
                bfr[ni][1] = Bs[cur][n][kk + 2*half + 1];
            }
            #pragma unroll
            for (int mi = 0; mi < 4; ++mi)
                #pragma unroll
                for (int ni = 0; ni < 2; ++ni)
                    acc[mi][ni] = __builtin_amdgcn_wmma_f32_16x16x4_f32(
                        false, afr[mi], false, bfr[ni], (short)0, acc[mi][ni], false, false);
        }
    }
    float* gC = pacts + (size_t)b * S * NPROC;
    #pragma unroll
    for (int mi = 0; mi < 4; ++mi)
        #pragma unroll
        for (int ni = 0; ni < 2; ++ni) {
            int ncol = n0 + wn * 32 + ni * 16 + lr;
            #pragma unroll
            for (int v = 0; v < 8; ++v) {
                int mrow = m0 + wm * 64 + mi * 16 + half * 8 + v;
                gC[(size_t)mrow * NPROC + ncol] = gelu_erf(acc[mi][ni][v]);
            }
        }
}

// ============================================================
// Kernel 4: process_scores[b][p] = mean over s of proc_acts[b][s][p]
// ============================================================
__global__ __launch_bounds__(256) void k_scores(const float* __restrict__ pa,
                                                float* __restrict__ sc) {
    int b = blockIdx.x >> 3;
    int p = ((blockIdx.x & 7) << 8) + threadIdx.x;
    const float* base = pa + (size_t)b * S * NPROC + p;
    float s = 0.f;
    for (int i = 0; i < S; ++i) s += base[(size_t)i * NPROC];
    sc[b * NPROC + p] = s * (1.0f / (float)S);
}

// ============================================================
// Kernel 5: top-1024 of 2048 process scores per batch (set semantics).
// ============================================================
__global__ __launch_bounds__(1024) void k_topk_proc(const float* __restrict__ scores,
                                                    int* __restrict__ sel) {
    __shared__ float sh_v[NPROC];
    __shared__ float sh_sort[NPROC];
    __shared__ float sh_scan[1024];
    const int b = blockIdx.x, tid = threadIdx.x;
    const float* sc = scores + b * NPROC;
    for (int n = tid; n < NPROC; n += 1024) { sh_v[n] = sc[n]; sh_sort[n] = sc[n]; }
    __syncthreads();
    for (int ksz = 2; ksz <= NPROC; ksz <<= 1) {
        for (int j = ksz >> 1; j > 0; j >>= 1) {
            for (int t = tid; t < NPROC; t += 1024) {
                int ixj = t ^ j;
                if (ixj > t) {
                    float a = sh_sort[t], c = sh_sort[ixj];
                    if (((t & ksz) == 0) ? (a < c) : (a > c)) { sh_sort[t] = c; sh_sort[ixj] = a; }
                }
            }
            __syncthreads();
        }
    }
    const float thresh = sh_sort[KPROC - 1];
    __syncthreads();
    const int base2 = tid * 2;
    int my_gt = 0, my_eq = 0;
    for (int u = 0; u < 2; ++u) {
        float l = sh_v[base2 + u];
        if (l > thresh) my_gt++; else if (l == thresh) my_eq++;
    }
    sh_scan[tid] = (float)my_eq; __syncthreads();
    for (int off = 1; off < 1024; off <<= 1) {
        float t = (tid >= off) ? sh_scan[tid - off] : 0.f;
        __syncthreads();
        sh_scan[tid] += t;
        __syncthreads();
    }
    int eq_before = (int)sh_scan[tid] - my_eq;
    __syncthreads();
    sh_scan[tid] = (float)my_gt; __syncthreads();
    for (int off = 512; off > 0; off >>= 1) {
        if (tid < off) sh_scan[tid] += sh_scan[tid + off];
        __syncthreads();
    }
    int total_gt = (int)sh_scan[0];
    __syncthreads();
    int need_eq = KPROC - total_gt;
    int flags[2]; int my_sel = 0; int er = eq_before;
    for (int u = 0; u < 2; ++u) {
        float l = sh_v[base2 + u];
        int f;
        if (l > thresh) f = 1;
        else if (l == thresh) { f = (er < need_eq) ? 1 : 0; er++; }
        else f = 0;
        flags[u] = f; my_sel += f;
    }
    sh_scan[tid] = (float)my_sel; __syncthreads();
    for (int off = 1; off < 1024; off <<= 1) {
        float t = (tid >= off) ? sh_scan[tid - off] : 0.f;
        __syncthreads();
        sh_scan[tid] += t;
        __syncthreads();
    }
    int pos = (int)sh_scan[tid] - my_sel;
    for (int u = 0; u < 2; ++u)
        if (flags[u]) sel[b * KPROC + pos++] = base2 + u;
}

// ============================================================
// Kernel 6: stage 3  out[b][s][d] = sum_p' pa[b][s][idx[p']] * po[idx[p']][d]
// C = A(MxK gathered cols, async b32) * B(KxN gathered rows, async b128).
// M=S, N=D, Kdim=KPROC.
// ============================================================
__global__ __launch_bounds__(256) void k_stage3(
    const float* __restrict__ pa, const float* __restrict__ po,
    const int* __restrict__ pidx, float* __restrict__ out) {
    __shared__ __align__(16) float As[2][128][36];
    __shared__ __align__(16) float Bs3[2][32][132];
    const int b  = blockIdx.z;
    const int m0 = blockIdx.x * 128;
    const int n0 = blockIdx.y * 128;
    const int tid = threadIdx.x;
    const int lane = tid & 31, wid = tid >> 5;
    const int wm = wid >> 2, wn = wid & 3;
    const int half = lane >> 4, lr = lane & 15;

    const float* gA = pa + (size_t)b * S * NPROC;
    const int*   pg = pidx + b * KPROC;

    v8f acc[4][2];
    v8f zero = {0.f,0.f,0.f,0.f,0.f,0.f,0.f,0.f};
    for (int i = 0; i < 4; ++i) for (int j = 0; j < 2; ++j) acc[i][j] = zero;

    auto issue = [&](int k0, int buf) {
        #pragma unroll
        for (int i = 0; i < 16; ++i) {          // gathered A columns
            int idx = tid + 256 * i;
            int row = idx >> 5, c = idx & 31;
            async_ld_b32(gA + (size_t)(m0 + row) * NPROC + pg[k0 + c], &As[buf][row][c]);
        }
        #pragma unroll
        for (int i = 0; i < 4; ++i) {           // gathered B rows (contiguous)
            int idx = tid + 256 * i;
            int row = idx >> 5, cg = idx & 31;
            async_ld_b128(po + (size_t)pg[k0 + row] * D + n0 + cg * 4, &Bs3[buf][row][cg * 4]);
        }
    };

    constexpr int NC = KPROC / 32;
    issue(0, 0);
    for (int c = 0; c < NC; ++c) {
        const int cur = c & 1;
        wait_async_lds();
        __syncthreads();
        if (c + 1 < NC) issue((c + 1) * 32, cur ^ 1);
        #pragma unroll
        for (int kk = 0; kk < 32; kk += 4) {
            v2f afr[4], bfr[2];
            #pragma unroll
            for (int mi = 0; mi < 4; ++mi) {
                int m = wm * 64 + mi * 16 + lr;
                afr[mi][0] = As[cur][m][kk + 2*half];
                afr[mi][1] = As[cur][m][kk + 2*half + 1];
            }
            #pragma unroll
            for (int ni = 0; ni < 2; ++ni) {
                int n = wn * 32 + ni * 16 + lr;
                bfr[ni][0] = Bs3[cur][kk + 2*half][n];
                bfr[ni][1] = Bs3[cur][kk + 2*half + 1][n];
            }
            #pragma unroll
            for (int mi = 0; mi < 4; ++mi)
                #pragma unroll
                for (int ni = 0; ni < 2; ++ni)
                    acc[mi][ni] = __builtin_amdgcn_wmma_f32_16x16x4_f32(
                        false, afr[mi], false, bfr[ni], (short)0, acc[mi][ni], false, false);
        }
    }
    float* gC = out + (size_t)b * S * D;
    #pragma unroll
    for (int mi = 0; mi < 4; ++mi)
        #pragma unroll
        for (int ni = 0; ni < 2; ++ni) {
            int ncol = n0 + wn * 32 + ni * 16 + lr;
            #pragma unroll
            for (int v = 0; v < 8; ++v) {
                int mrow = m0 + wm * 64 + mi * 16 + half * 8 + v;
                gC[(size_t)mrow * D + ncol] = acc[mi][ni][v];
            }
        }
}

// ============================================================
extern "C" void kernel_launch(void* const* d_in, const int* in_sizes, int n_in,
                              void* d_out, int out_size, void* d_ws, size_t ws_size,
                              hipStream_t stream) {
    const float* x    = (const float*)d_in[0];
    // d_in[1] = k_input (2048), d_in[2] = k_process (1024): fixed by reference
    const float* W1   = (const float*)d_in[3];
    const float* b1   = (const float*)d_in[4];
    const float* lng  = (const float*)d_in[5];
    const float* lnb  = (const float*)d_in[6];
    const float* W2   = (const float*)d_in[7];
    const float* b2   = (const float*)d_in[8];
    const float* nk   = (const float*)d_in[9];
    const float* patt = (const float*)d_in[10];
    const float* pw   = (const float*)d_in[11];
    const float* po   = (const float*)d_in[12];
    float* out = (float*)d_out;

    // workspace carve-up
    float* ws        = (float*)d_ws;
    float* gctx      = ws;                         // B*D
    float* selw      = gctx + B * D;               // B*KIN
    float* pscore    = selw + B * KIN;             // B*NPROC
    int*   selidx    = (int*)(pscore + B * NPROC); // B*KIN ints
    int*   selproc   = selidx + B * KIN;           // B*KPROC ints
    float* selin_buf = (float*)(selproc + B * KPROC);          // B*S*KIN
    float* pacts     = selin_buf + (size_t)B * S * KIN;        // B*S*NPROC

    k_gctx     <<<dim3(B * (D / 256)), dim3(256),  0, stream>>>(x, gctx);
    k_router   <<<dim3(B),             dim3(1024), 0, stream>>>(gctx, W1, b1, lng, lnb, W2, b2, nk, selidx, selw);
    k_stage1   <<<dim3(S/128, KIN/128, B),   dim3(256), 0, stream>>>(x, patt, selidx, selw, selin_buf);
    k_stage2   <<<dim3(S/128, NPROC/128, B), dim3(256), 0, stream>>>(selin_buf, pw, selidx, pacts);
    k_scores   <<<dim3(B * (NPROC / 256)), dim3(256),  0, stream>>>(pacts, pscore);
    k_topk_proc<<<dim3(B),                 dim3(1024), 0, stream>>>(pscore, selproc);
    k_stage3   <<<dim3(S/128, D/128, B),   dim3(256), 0, stream>>>(pacts, po, selproc, out);
}